// FinalNet_12524124636046
// MI455X (gfx1250) — compile-verified
//
#include <hip/hip_runtime.h>

// ---------------------------------------------------------------------------
// Problem constants (from the reference)
// ---------------------------------------------------------------------------
#define NU    60001        // U + 1
#define NI    30001        // I + 1
#define DD    64
#define DALLC 192
#define BB    1024
#define KN    16
#define KK1   50
#define KK2   50
#define FKC   20
#define KK3   30
#define SSC   48
#define LRELU 0.01f
#define TW    68           // padded LDS row stride (floats) for 64-wide tiles
#define TW2   52           // padded LDS row stride for 48-wide tiles
#define PW    160          // paired-W LDS pair-row stride (floats): 128 data + 32
                           //   -> consecutive pair rows shift banks by 32, so the
                           //      two half-waves (pairs p, p+1) use disjoint banks

typedef float v2f __attribute__((ext_vector_type(2)));
typedef float v4f __attribute__((ext_vector_type(4)));
typedef float v8f __attribute__((ext_vector_type(8)));

// f32 WMMA helper: D(16x16) += A(16x4) * B(4x16), CDNA5 V_WMMA_F32_16X16X4_F32
__device__ __forceinline__ v8f wmma_f32(v2f a, v2f b, v8f c) {
    return __builtin_amdgcn_wmma_f32_16x16x4_f32(
        /*neg_a=*/false, a, /*neg_b=*/false, b,
        /*c_mod=*/(short)0, c, /*reuse_a=*/false, /*reuse_b=*/false);
}

// ---------------------------------------------------------------------------
// Kernel 1: one GraphSAGE layer for a whole node table.
//   out[n, :] = l2norm( leaky_relu( h[n] @ Wself + mean_k h[neigh[n,k]] @ Wneigh ) )
// One wave owns a 16-row tile; K stacked to 128: A=[h_self | nm],
// B=[Wself ; Wneigh] staged PAIRED in LDS so each B fragment is one ds_load_b64.
// K-loop split by A source so self-rows stay global_load (no generic/flat).
// ---------------------------------------------------------------------------
__global__ __launch_bounds__(128) void graph_layer_kernel(
    const float* __restrict__ h_in, int in_stride,
    const int*   __restrict__ neigh,
    const float* __restrict__ Wself,   // [64][64] row-major (k, n)
    const float* __restrict__ Wneigh,  // [64][64]
    float* __restrict__ out, int out_stride, int N)
{
    __shared__ __align__(16) float sWp[64 * PW];      // paired stacked weights
    __shared__ __align__(16) float sT[4][16 * TW];    // per-wave tile (nm, then D)
    __shared__ float sRn[4][16];

    const int tid  = threadIdx.x;
    const int wave = tid >> 5;
    const int lane = tid & 31;
    const int l15  = lane & 15;
    const int hi   = lane >> 4;                       // half-wave select

    // stage paired: sWp[p][2n+j] = Wstacked[2p+j][n], rows 0..63 self, 64..127 neigh
    for (int i = tid; i < 64 * 64; i += 128) {
        int r = i >> 6, c = i & 63;
        sWp[(r >> 1) * PW + 2 * c + (r & 1)]        = Wself[i];
        sWp[((r >> 1) + 32) * PW + 2 * c + (r & 1)] = Wneigh[i];
    }
    __syncthreads();

    const int tileRow = (blockIdx.x * 4 + wave) * 16;
    float* nm = &sT[wave][0];

    // ---- neighbor mean into nm[16][TW] (half-wave per row, float4 per lane)
    const int qc = l15;                               // float4 column 0..15
    for (int i = 0; i < 8; ++i) {
        int r = hi + 2 * i;
        int g = tileRow + r; if (g >= N) g = N - 1;
        const int* nrow = neigh + (size_t)g * KN;
        v4f acc = {0.f, 0.f, 0.f, 0.f};
        #pragma unroll
        for (int k = 0; k < KN; ++k) {
            int idx = nrow[k];
            acc += *(const v4f*)(h_in + (size_t)idx * in_stride + 4 * qc);
        }
        acc *= (1.0f / (float)KN);
        *(v4f*)&nm[r * TW + 4 * qc] = acc;
    }
    __syncthreads();

    // ---- WMMA: D[16x64] = A[16x128] x B[128x64]
    v8f acc0 = {}, acc1 = {}, acc2 = {}, acc3 = {};
    int rowg = tileRow + l15; if (rowg >= N) rowg = N - 1;
    const float* selfRow = h_in + (size_t)rowg * in_stride;

    // phase 1: A from global (self rows), K = 0..63
    for (int kk = 0; kk < 64; kk += 4) {
        v2f a = *(const v2f*)(selfRow + kk + 2 * hi);
        const float* wp = &sWp[((kk >> 1) + hi) * PW + 2 * l15];
        v2f b;
        b = *(const v2f*)(wp +  0); acc0 = wmma_f32(a, b, acc0);
        b = *(const v2f*)(wp + 32); acc1 = wmma_f32(a, b, acc1);
        b = *(const v2f*)(wp + 64); acc2 = wmma_f32(a, b, acc2);
        b = *(const v2f*)(wp + 96); acc3 = wmma_f32(a, b, acc3);
    }
    // phase 2: A from LDS (neighbor means), K = 64..127
    for (int kk = 64; kk < 128; kk += 4) {
        v2f a = *(const v2f*)&nm[l15 * TW + (kk - 64) + 2 * hi];
        const float* wp = &sWp[((kk >> 1) + hi) * PW + 2 * l15];
        v2f b;
        b = *(const v2f*)(wp +  0); acc0 = wmma_f32(a, b, acc0);
        b = *(const v2f*)(wp + 32); acc1 = wmma_f32(a, b, acc1);
        b = *(const v2f*)(wp + 64); acc2 = wmma_f32(a, b, acc2);
        b = *(const v2f*)(wp + 96); acc3 = wmma_f32(a, b, acc3);
    }

    // ---- leaky ReLU, stage D in LDS (reuse nm tile)
    float* Dt = nm;
    {
        v8f* accs[4] = {&acc0, &acc1, &acc2, &acc3};
        #pragma unroll
        for (int bI = 0; bI < 4; ++bI) {
            #pragma unroll
            for (int e = 0; e < 8; ++e) {
                float x = (*accs[bI])[e];
                x = (x < 0.f) ? LRELU * x : x;
                Dt[(hi * 8 + e) * TW + bI * 16 + l15] = x;  // row = e + 8*hi
            }
        }
    }
    __syncthreads();

    // ---- per-row L2 norm
    if (lane < 16) {
        float s = 0.f;
        #pragma unroll 8
        for (int c = 0; c < 64; ++c) { float v = Dt[lane * TW + c]; s += v * v; }
        sRn[wave][lane] = 1.0f / fmaxf(sqrtf(s), 1e-12f);
    }
    __syncthreads();

    // ---- write out (coalesced float4)
    for (int i = 0; i < 8; ++i) {
        int r = hi + 2 * i;
        int g = tileRow + r;
        if (g < N) {
            v4f v = *(v4f*)&Dt[r * TW + 4 * qc];
            v *= sRn[wave][r];
            *(v4f*)(out + (size_t)g * out_stride + 4 * qc) = v;
        }
    }
}

// ---------------------------------------------------------------------------
// Kernel 2: finalize an embedding table (float4-vectorized).
//   emb[:,0:64] = base64 ; if halfMode: emb[:,64:192] = 0.5*(emb[:,64:192]+other128)
//   emb[0,:] = 0
// ---------------------------------------------------------------------------
__global__ void combine_kernel(float* __restrict__ emb,
                               const float* __restrict__ base64,
                               const float* __restrict__ other128,
                               int N, int halfMode)
{
    size_t total = (size_t)N * (DALLC / 4);
    for (size_t i = (size_t)blockIdx.x * blockDim.x + threadIdx.x; i < total;
         i += (size_t)gridDim.x * blockDim.x) {
        int n = (int)(i / (DALLC / 4));
        int c = 4 * (int)(i % (DALLC / 4));
        v4f* dst = (v4f*)(emb + (size_t)n * DALLC + c);
        if (n == 0) {
            v4f z = {0.f, 0.f, 0.f, 0.f};
            *dst = z;
        } else if (c < 64) {
            *dst = *(const v4f*)(base64 + (size_t)n * 64 + c);
        } else if (halfMode) {
            v4f v = *dst;
            v4f o = *(const v4f*)(other128 + (size_t)n * 128 + (c - 64));
            *dst = 0.5f * (v + o);
        }
    }
}

// ---------------------------------------------------------------------------
// Kernel 3: gathers + elementwise-mul + max fusions. One block per batch row,
// thread = embedding column (coalesced row reads from L2-resident tables).
// ---------------------------------------------------------------------------
__global__ __launch_bounds__(192) void gather_max_kernel(
    const int* __restrict__ user, const int* __restrict__ cand,
    const int* __restrict__ item_users, const int* __restrict__ user_items,
    const int* __restrict__ friends_items,
    const float* __restrict__ UG, const float* __restrict__ IS,
    float* __restrict__ uglobal, float* __restrict__ isglobal,
    float* __restrict__ ilocal, float* __restrict__ usl,
    float* __restrict__ femb)
{
    const int b = blockIdx.x, d = threadIdx.x;
    const float ug = UG[(size_t)user[b] * DALLC + d];
    const float ig = IS[(size_t)cand[b] * DALLC + d];
    uglobal[b * DALLC + d]  = ug;
    isglobal[b * DALLC + d] = ig;

    float m = -3.402823466e38f;
    for (int k = 0; k < KK1; ++k) {
        int idx = item_users[b * KK1 + k];
        m = fmaxf(m, UG[(size_t)idx * DALLC + d] * ug);
    }
    ilocal[b * DALLC + d] = m;

    m = -3.402823466e38f;
    for (int k = 0; k < KK2; ++k) {
        int idx = user_items[b * KK2 + k];
        m = fmaxf(m, IS[(size_t)idx * DALLC + d] * ig);
    }
    usl[b * DALLC + d] = m;

    for (int f = 0; f < FKC; ++f) {
        m = -3.402823466e38f;
        for (int k = 0; k < KK3; ++k) {
            int idx = friends_items[(b * FKC + f) * KK3 + k];
            m = fmaxf(m, IS[(size_t)idx * DALLC + d] * ig);
        }
        femb[(size_t)(b * FKC + f) * DALLC + d] = m;
    }
}

// ---------------------------------------------------------------------------
// Prep: pack a [384,48] weight matrix into paired layout:
//   dst[p*96 + 2n + j] = src[(2p+j)*48 + n]   (p<192, n<48, j<2)
// so the row-MLP's B fragment is a single aligned global_load_b64.
// ---------------------------------------------------------------------------
__global__ void pack_w1_kernel(const float* __restrict__ src, float* __restrict__ dst)
{
    int i = blockIdx.x * blockDim.x + threadIdx.x;
    if (i < 2 * DALLC * SSC) {
        int k = i / SSC, n = i % SSC;
        dst[(k >> 1) * (2 * SSC) + 2 * n + (k & 1)] = src[i];
    }
}

// ---------------------------------------------------------------------------
// Kernel 4 (WMMA): generic row scorer
//   out[r] = leaky_relu( concat(srcA[r/divA], srcB[r]) @ W1 ) . w2
// W1p is the PAIRED [192 pairs][96] layout produced by pack_w1_kernel.
// ---------------------------------------------------------------------------
__global__ __launch_bounds__(128) void rowmlp_kernel(
    const float* __restrict__ srcA, int divA,
    const float* __restrict__ srcB,
    const float* __restrict__ W1p,  // paired [192][96]
    const float* __restrict__ w2,   // [48]
    float* __restrict__ out, int nRows)
{
    __shared__ __align__(16) float sD[4][16 * TW2];
    __shared__ float sw2[SSC];

    const int tid  = threadIdx.x;
    const int wave = tid >> 5;
    const int lane = tid & 31;
    const int l15  = lane & 15;
    const int hi   = lane >> 4;

    if (tid < SSC) sw2[tid] = w2[tid];
    __syncthreads();

    const int tileRow = (blockIdx.x * 4 + wave) * 16;
    int rowg = tileRow + l15; if (rowg >= nRows) rowg = nRows - 1;
    const float* rA = srcA + (size_t)(rowg / divA) * DALLC;
    const float* rB = srcB + (size_t)rowg * DALLC;

    v8f a0 = {}, a1 = {}, a2 = {};
    // phase 1: A from srcA rows, K = 0..191
    for (int kk = 0; kk < DALLC; kk += 4) {
        v2f a = *(const v2f*)(rA + kk + 2 * hi);
        const float* wp = W1p + (size_t)((kk >> 1) + hi) * (2 * SSC) + 2 * l15;
        v2f b;
        b = *(const v2f*)(wp +  0); a0 = wmma_f32(a, b, a0);
        b = *(const v2f*)(wp + 32); a1 = wmma_f32(a, b, a1);
        b = *(const v2f*)(wp + 64); a2 = wmma_f32(a, b, a2);
    }
    // phase 2: A from srcB rows, K = 192..383
    for (int kk = DALLC; kk < 2 * DALLC; kk += 4) {
        v2f a = *(const v2f*)(rB + (kk - DALLC) + 2 * hi);
        const float* wp = W1p + (size_t)((kk >> 1) + hi) * (2 * SSC) + 2 * l15;
        v2f b;
        b = *(const v2f*)(wp +  0); a0 = wmma_f32(a, b, a0);
        b = *(const v2f*)(wp + 32); a1 = wmma_f32(a, b, a1);
        b = *(const v2f*)(wp + 64); a2 = wmma_f32(a, b, a2);
    }

    float* Dt = &sD[wave][0];
    {
        v8f* accs[3] = {&a0, &a1, &a2};
        #pragma unroll
        for (int bI = 0; bI < 3; ++bI) {
            #pragma unroll
            for (int e = 0; e < 8; ++e) {
                float x = (*accs[bI])[e];
                x = (x < 0.f) ? LRELU * x : x;
                Dt[(hi * 8 + e) * TW2 + bI * 16 + l15] = x;
            }
        }
    }
    __syncthreads();

    if (lane < 16) {
        int g = tileRow + lane;
        if (g < nRows) {
            float s = 0.f;
            #pragma unroll 8
            for (int c = 0; c < SSC; ++c) s += Dt[lane * TW2 + c] * sw2[c];
            out[g] = s;
        }
    }
}

// ---------------------------------------------------------------------------
// Kernel 5: masked softmax over friends + weighted sum + user_local
// ---------------------------------------------------------------------------
__global__ __launch_bounds__(192) void attn_softmax_kernel(
    const float* __restrict__ logits, const int* __restrict__ flens,
    const float* __restrict__ femb, const float* __restrict__ usl,
    float* __restrict__ ulocal)
{
    __shared__ float sa[FKC];
    const int b = blockIdx.x, t = threadIdx.x;
    if (t == 0) {
        int L = flens[b]; if (L < 1) L = 1;
        float tmp[FKC];
        float mx = -1e30f;
        #pragma unroll
        for (int f = 0; f < FKC; ++f) {
            float v = (f < L) ? logits[b * FKC + f] : -1e9f;
            tmp[f] = v; mx = fmaxf(mx, v);
        }
        float s = 0.f;
        #pragma unroll
        for (int f = 0; f < FKC; ++f) { float e = __expf(tmp[f] - mx); tmp[f] = e; s += e; }
        float inv = 1.0f / s;
        #pragma unroll
        for (int f = 0; f < FKC; ++f) sa[f] = tmp[f] * inv;
    }
    __syncthreads();
    float acc = 0.f;
    #pragma unroll
    for (int f = 0; f < FKC; ++f)
        acc += sa[f] * femb[(size_t)(b * FKC + f) * DALLC + t];
    ulocal[b * DALLC + t] = 0.5f * (usl[b * DALLC + t] + acc);
}

// ---------------------------------------------------------------------------
// Kernel 6: final average of the three predictor scores
// ---------------------------------------------------------------------------
__global__ void final_kernel(const float* __restrict__ s1, const float* __restrict__ s2,
                             const float* __restrict__ s3, float* __restrict__ out, int n)
{
    int i = blockIdx.x * blockDim.x + threadIdx.x;
    if (i < n) out[i] = (s1[i] + s2[i] + s3[i]) * (1.0f / 3.0f);
}

// ---------------------------------------------------------------------------
extern "C" void kernel_launch(void* const* d_in, const int* in_sizes, int n_in,
                              void* d_out, int out_size, void* d_ws, size_t ws_size,
                              hipStream_t stream)
{
    (void)in_sizes; (void)n_in; (void)out_size; (void)ws_size;

    const int*   user        = (const int*)d_in[0];
    const int*   candidate   = (const int*)d_in[1];
    const int*   soc_neigh   = (const int*)d_in[2];
    const int*   usim_neigh  = (const int*)d_in[3];
    const int*   isim_neigh  = (const int*)d_in[4];
    const int*   item_users  = (const int*)d_in[5];
    const int*   user_items  = (const int*)d_in[6];
    const int*   friends_it  = (const int*)d_in[7];
    const int*   friends_len = (const int*)d_in[8];
    const float* user_emb    = (const float*)d_in[9];
    const float* item_emb    = (const float*)d_in[10];
    const float* Wsoc_self   = (const float*)d_in[11];
    const float* Wsoc_neigh  = (const float*)d_in[12];
    const float* Wsim_self   = (const float*)d_in[13];
    const float* Wsim_neigh  = (const float*)d_in[14];
    const float* Wit_self    = (const float*)d_in[15];
    const float* Wit_neigh   = (const float*)d_in[16];
    const float* att_w1      = (const float*)d_in[17];
    const float* att_w2      = (const float*)d_in[18];
    const float* p1_w1       = (const float*)d_in[19];
    const float* p2_w1       = (const float*)d_in[20];
    const float* p3_w1       = (const float*)d_in[21];
    const float* p1_w2       = (const float*)d_in[22];
    const float* p2_w2       = (const float*)d_in[23];
    const float* p3_w2       = (const float*)d_in[24];

    float* ws = (float*)d_ws;
    size_t off = 0;
    float* UG   = ws + off; off += (size_t)NU * DALLC;   // user_global_emb [NU,192]
    float* IS   = ws + off; off += (size_t)NI * DALLC;   // i_sim           [NI,192]
    float* E2   = ws + off; off += (size_t)NU * 128;     // sim layers 1,2  [NU,128]
    float* FEMB = ws + off; off += (size_t)BB * FKC * DALLC;
    float* UGB  = ws + off; off += (size_t)BB * DALLC;   // user_global
    float* ISG  = ws + off; off += (size_t)BB * DALLC;   // item_sim_global
    float* ILOC = ws + off; off += (size_t)BB * DALLC;   // item_local
    float* USL  = ws + off; off += (size_t)BB * DALLC;   // user_sim_local
    float* ULOC = ws + off; off += (size_t)BB * DALLC;   // user_local
    float* LOGI = ws + off; off += (size_t)BB * FKC;     // attention logits
    float* S1   = ws + off; off += BB;
    float* S2   = ws + off; off += BB;
    float* S3   = ws + off; off += BB;
    float* ATTP = ws + off; off += (size_t)2 * DALLC * SSC;  // paired W1 copies
    float* P1P  = ws + off; off += (size_t)2 * DALLC * SSC;
    float* P2P  = ws + off; off += (size_t)2 * DALLC * SSC;
    float* P3P  = ws + off; off += (size_t)2 * DALLC * SSC;

    const dim3 blk(128);
    const int gU = (NU + 63) / 64, gI = (NI + 63) / 64;
    const int WL = DD * DD;  // per-layer weight stride
    const int gPack = (2 * DALLC * SSC + 255) / 256;

    // pack predictor/attention W1 into paired layout (tiny, runs up front)
    pack_w1_kernel<<<gPack, 256, 0, stream>>>(att_w1, ATTP);
    pack_w1_kernel<<<gPack, 256, 0, stream>>>(p1_w1, P1P);
    pack_w1_kernel<<<gPack, 256, 0, stream>>>(p2_w1, P2P);
    pack_w1_kernel<<<gPack, 256, 0, stream>>>(p3_w1, P3P);

    // social graph: layers -> UG[:,64:128], UG[:,128:192]
    graph_layer_kernel<<<gU, blk, 0, stream>>>(user_emb, DD, soc_neigh,
        Wsoc_self, Wsoc_neigh, UG + 64, DALLC, NU);
    graph_layer_kernel<<<gU, blk, 0, stream>>>(UG + 64, DALLC, soc_neigh,
        Wsoc_self + WL, Wsoc_neigh + WL, UG + 128, DALLC, NU);
    // user-sim graph: layers -> E2[:,0:64], E2[:,64:128]
    graph_layer_kernel<<<gU, blk, 0, stream>>>(user_emb, DD, usim_neigh,
        Wsim_self, Wsim_neigh, E2, 128, NU);
    graph_layer_kernel<<<gU, blk, 0, stream>>>(E2, 128, usim_neigh,
        Wsim_self + WL, Wsim_neigh + WL, E2 + 64, 128, NU);
    // UG = 0.5*(soc + sim), layer0 = user_emb, row0 = 0
    combine_kernel<<<2048, 256, 0, stream>>>(UG, user_emb, E2, NU, 1);

    // item-sim graph: layers -> IS[:,64:128], IS[:,128:192]
    graph_layer_kernel<<<gI, blk, 0, stream>>>(item_emb, DD, isim_neigh,
        Wit_self, Wit_neigh, IS + 64, DALLC, NI);
    graph_layer_kernel<<<gI, blk, 0, stream>>>(IS + 64, DALLC, isim_neigh,
        Wit_self + WL, Wit_neigh + WL, IS + 128, DALLC, NI);
    combine_kernel<<<1024, 256, 0, stream>>>(IS, item_emb, nullptr, NI, 0);

    // gathers + max fusions
    gather_max_kernel<<<BB, 192, 0, stream>>>(user, candidate, item_users,
        user_items, friends_it, UG, IS, UGB, ISG, ILOC, USL, FEMB);

    // attention logits over friends (WMMA row-MLP), softmax, user_local
    rowmlp_kernel<<<(BB * FKC + 63) / 64, blk, 0, stream>>>(USL, FKC, FEMB,
        ATTP, att_w2, LOGI, BB * FKC);
    attn_softmax_kernel<<<BB, 192, 0, stream>>>(LOGI, friends_len, FEMB, USL, ULOC);

    // three predictors (WMMA row-MLP) + final average
    rowmlp_kernel<<<(BB + 63) / 64, blk, 0, stream>>>(UGB,  1, ISG,  P1P, p1_w2, S1, BB);
    rowmlp_kernel<<<(BB + 63) / 64, blk, 0, stream>>>(ULOC, 1, ISG,  P2P, p2_w2, S2, BB);
    rowmlp_kernel<<<(BB + 63) / 64, blk, 0, stream>>>(UGB,  1, ILOC, P3P, p3_w2, S3, BB);
    final_kernel<<<(BB + 255) / 256, 256, 0, stream>>>(S1, S2, S3, (float*)d_out, BB);
}